// GAT_31456340476248
// MI455X (gfx1250) — compile-verified
//
#include <hip/hip_runtime.h>
#include <hip/hip_bf16.h>
#include <math.h>

typedef __attribute__((ext_vector_type(2))) float v2f;
typedef __attribute__((ext_vector_type(8))) float v8f;

#define NEG_SLOPE 0.2f

// Monotone float <-> uint mapping so atomicMax(uint) implements float max.
__device__ __forceinline__ unsigned flipf(float f) {
  unsigned u = __float_as_uint(f);
  return (u & 0x80000000u) ? ~u : (u | 0x80000000u);
}
__device__ __forceinline__ float unflipf(unsigned k) {
  unsigned u = (k & 0x80000000u) ? (k ^ 0x80000000u) : ~k;
  return __uint_as_float(u);
}

__global__ void zero_kernel(float* __restrict__ p, long long n) {
  long long i = (long long)blockIdx.x * blockDim.x + threadIdx.x;
  if (i < n) p[i] = 0.f;
}

// C[M,NC] = A[M,128] * B[128,NC], fp32 via V_WMMA_F32_16X16X4_F32.
// Block = 256 threads = 8 waves; covers 32 rows x 64 cols (2 row tiles x 4 col tiles).
// A panel and B panel staged in LDS; A rows padded (LDA=K+4) to avoid bank conflicts.
template<int NC>
__global__ __launch_bounds__(256) void gemm_f32_wmma(const float* __restrict__ A,
                                                     const float* __restrict__ B,
                                                     float* __restrict__ C, int M) {
  constexpr int K = 128;
  constexpr int CG = 64;            // columns per block
  constexpr int LDA = K + 4;
  constexpr int colGroups = NC / CG;
  __shared__ float Wlds[K * CG];    // 32 KB
  __shared__ float Alds[32 * LDA];  // ~16.9 KB
  int cg = blockIdx.x % colGroups;
  int rg = blockIdx.x / colGroups;

  // stage B panel [K][CG]
  for (int i = threadIdx.x; i < K * (CG / 4); i += 256) {
    int k = i / (CG / 4), c4 = i % (CG / 4);
    *(float4*)&Wlds[k * CG + c4 * 4] =
        *(const float4*)&B[(size_t)k * NC + cg * CG + c4 * 4];
  }
  // stage A panel (32 rows x K)
  for (int i = threadIdx.x; i < 32 * (K / 4); i += 256) {
    int r = i / (K / 4), c4 = i % (K / 4);
    int grow = rg * 32 + r;
    float4 v = make_float4(0.f, 0.f, 0.f, 0.f);
    if (grow < M) v = *(const float4*)&A[(size_t)grow * K + c4 * 4];
    *(float4*)&Alds[r * LDA + c4 * 4] = v;
  }
  __syncthreads();

  int wave = threadIdx.x >> 5;
  int lane = threadIdx.x & 31;
  int half = lane >> 4;             // lanes 16-31 hold K+2/K+3 (A) and rows k+2/k+3 (B), M+8 (C)
  int lrow = lane & 15;
  int rt = wave >> 2, ct = wave & 3;
  int grow0 = rg * 32 + rt * 16;
  if (grow0 >= M) return;           // wave-uniform: EXEC stays all-ones for WMMA

  const float* Ab = &Alds[(rt * 16 + lrow) * LDA + 2 * half];
  const float* Bb = &Wlds[(2 * half) * CG + ct * 16 + lrow];
  v8f acc = {};
  #pragma unroll
  for (int k0 = 0; k0 < K; k0 += 4) {
    v2f a, b;
    a.x = Ab[k0];        a.y = Ab[k0 + 1];
    b.x = Bb[k0 * CG];   b.y = Bb[k0 * CG + CG];
    acc = __builtin_amdgcn_wmma_f32_16x16x4_f32(false, a, false, b,
                                                (short)0, acc, false, false);
  }
  float* Cp = &C[(size_t)(grow0 + 8 * half) * NC + cg * CG + ct * 16 + lrow];
  #pragma unroll
  for (int j = 0; j < 8; ++j) Cp[(size_t)j * NC] = acc[j];
}

// Per-(node,head) attention logits: asrc[n,h]=dot(h[n,h,:],a_src[h,:]), same for dst. C=64.
template<int H>
__global__ void node_alpha_kernel(const float* __restrict__ h,
                                  const float* __restrict__ aw_src,
                                  const float* __restrict__ aw_dst,
                                  float* __restrict__ asrc, float* __restrict__ adst, int N) {
  int idx = blockIdx.x * blockDim.x + threadIdx.x;
  if (idx >= N * H) return;
  int hh = idx % H;
  const float* hp = h + (size_t)idx * 64;
  const float* ap = aw_src + hh * 64;
  const float* bp = aw_dst + hh * 64;
  float sa = 0.f, sb = 0.f;
  #pragma unroll
  for (int c = 0; c < 64; c += 4) {
    float4 v = *(const float4*)(hp + c);
    float4 u = *(const float4*)(ap + c);
    float4 t = *(const float4*)(bp + c);
    sa += v.x * u.x + v.y * u.y + v.z * u.z + v.w * u.w;
    sb += v.x * t.x + v.y * t.y + v.z * t.z + v.w * t.w;
  }
  asrc[idx] = sa;
  adst[idx] = sb;
}

// Pass 1: segment max of leaky_relu(asrc[src]+adst[dst]) per (dst, head).
template<int H>
__global__ void edge_max_kernel(const int* __restrict__ srcE, const int* __restrict__ dstE,
                                const float* __restrict__ asrc, const float* __restrict__ adst,
                                unsigned* __restrict__ m, int E, int Etot) {
  int idx = blockIdx.x * blockDim.x + threadIdx.x;
  if (idx >= Etot * H) return;
  int e = idx / H, hh = idx % H;
  int s = (e < E) ? srcE[e] : (e - E);   // implicit self-loops for e >= E
  int d = (e < E) ? dstE[e] : (e - E);
  float v = asrc[s * H + hh] + adst[d * H + hh];
  v = (v > 0.f) ? v : NEG_SLOPE * v;
  atomicMax(&m[d * H + hh], flipf(v));
}

// Pass 2: e = exp(v - m[dst]); store per-edge e, accumulate segment sum.
template<int H>
__global__ void edge_expsum_kernel(const int* __restrict__ srcE, const int* __restrict__ dstE,
                                   const float* __restrict__ asrc, const float* __restrict__ adst,
                                   const unsigned* __restrict__ m, float* __restrict__ ssum,
                                   float* __restrict__ eout, int E, int Etot) {
  int idx = blockIdx.x * blockDim.x + threadIdx.x;
  if (idx >= Etot * H) return;
  int e = idx / H, hh = idx % H;
  int s = (e < E) ? srcE[e] : (e - E);
  int d = (e < E) ? dstE[e] : (e - E);
  float v = asrc[s * H + hh] + adst[d * H + hh];
  v = (v > 0.f) ? v : NEG_SLOPE * v;
  float ex = __expf(v - unflipf(m[d * H + hh]));
  eout[idx] = ex;                        // idx == e*H+hh
  atomicAdd(&ssum[d * H + hh], ex);
}

// Pass 3: agg[dst,h,:] += (e/s[dst,h]) * h[src,h,:]; one float4 of channels per thread.
template<int H>
__global__ void edge_agg_kernel(const int* __restrict__ srcE, const int* __restrict__ dstE,
                                const float* __restrict__ eexp, const float* __restrict__ ssum,
                                const float* __restrict__ h, float* __restrict__ agg,
                                int E, int Etot) {
  constexpr int G = H * 64 / 4;          // float4 groups per edge (32 for H=2, 16 for H=1)
  long long idx = (long long)blockIdx.x * blockDim.x + threadIdx.x;
  if (idx >= (long long)Etot * G) return;
  int g = (int)(idx % G);
  int e = (int)(idx / G);
  int s = (e < E) ? srcE[e] : (e - E);
  int d = (e < E) ? dstE[e] : (e - E);
  int hh = g >> 4;                       // (g*4)/64
  int c = (g & 15) * 4;
  float alpha = eexp[e * H + hh] / ssum[d * H + hh];
  float4 hv = *(const float4*)&h[((size_t)s * H + hh) * 64 + c];
  float* ap = &agg[((size_t)d * H + hh) * 64 + c];
  atomicAdd(ap + 0, alpha * hv.x);
  atomicAdd(ap + 1, alpha * hv.y);
  atomicAdd(ap + 2, alpha * hv.z);
  atomicAdd(ap + 3, alpha * hv.w);
}

__global__ void bias_relu_kernel(float* __restrict__ a, const float* __restrict__ b,
                                 long long n, int Fmask) {
  long long i = (long long)blockIdx.x * blockDim.x + threadIdx.x;
  if (i >= n) return;
  float v = a[i] + b[(int)(i & Fmask)];
  a[i] = v > 0.f ? v : 0.f;
}

__global__ void bias_sigmoid_kernel(const float* __restrict__ a, const float* __restrict__ b,
                                    float* __restrict__ out, long long n, int Fmask) {
  long long i = (long long)blockIdx.x * blockDim.x + threadIdx.x;
  if (i >= n) return;
  float v = a[i] + b[(int)(i & Fmask)];
  out[i] = 1.f / (1.f + __expf(-v));
}

extern "C" void kernel_launch(void* const* d_in, const int* in_sizes, int n_in,
                              void* d_out, int out_size, void* d_ws, size_t ws_size,
                              hipStream_t stream) {
  const float* x   = (const float*)d_in[0];
  const int*   ei  = (const int*)d_in[1];
  const float* W1  = (const float*)d_in[2];
  const float* as1 = (const float*)d_in[3];
  const float* ad1 = (const float*)d_in[4];
  const float* b1  = (const float*)d_in[5];
  const float* W2  = (const float*)d_in[6];
  const float* as2 = (const float*)d_in[7];
  const float* ad2 = (const float*)d_in[8];
  const float* b2  = (const float*)d_in[9];
  float* out = (float*)d_out;

  const int N = in_sizes[0] / 128;
  const int E = in_sizes[1] / 2;
  const int Etot = E + N;
  const int* srcE = ei;
  const int* dstE = ei + E;

  // workspace layout (floats)
  float* w = (float*)d_ws;
  size_t o = 0;
  float* h1    = w + o; o += (size_t)N * 128;   // layer-1 features; reused for layer 2
  float* agg1  = w + o; o += (size_t)N * 128;   // layer-1 aggregate -> relu'd in place
  float* asrc1 = w + o; o += (size_t)N * 2;
  float* adst1 = w + o; o += (size_t)N * 2;
  unsigned* m1 = (unsigned*)(w + o); o += (size_t)N * 2;  // m1,s1 contiguous
  float* s1    = w + o; o += (size_t)N * 2;
  float* asrc2 = w + o; o += (size_t)N;
  float* adst2 = w + o; o += (size_t)N;
  unsigned* m2 = (unsigned*)(w + o); o += (size_t)N;      // m2,s2 contiguous
  float* s2    = w + o; o += (size_t)N;
  float* e1    = w + o; o += (size_t)Etot * 2;
  float* e2    = w + o; o += (size_t)Etot;
  float* h2f   = h1;                       // layer-2 features [N,64] (h1 is dead by then)
  float* agg2  = h1 + (size_t)N * 64;      // layer-2 aggregate [N,64]
  (void)o; (void)ws_size; (void)n_in; (void)out_size;

  const int TB = 256;
  auto blk = [](long long n, int tb) { return (unsigned)((n + tb - 1) / tb); };
  const int rowGroups = (N + 31) / 32;

  // ---------------- layer 1 (H=2, C=64) ----------------
  zero_kernel<<<blk((long long)N * 128, TB), TB, 0, stream>>>(agg1, (long long)N * 128);
  zero_kernel<<<blk((long long)N * 4, TB), TB, 0, stream>>>((float*)m1, (long long)N * 4);
  gemm_f32_wmma<128><<<dim3(rowGroups * 2), TB, 0, stream>>>(x, W1, h1, N);
  node_alpha_kernel<2><<<blk((long long)N * 2, TB), TB, 0, stream>>>(h1, as1, ad1, asrc1, adst1, N);
  edge_max_kernel<2><<<blk((long long)Etot * 2, TB), TB, 0, stream>>>(srcE, dstE, asrc1, adst1, m1, E, Etot);
  edge_expsum_kernel<2><<<blk((long long)Etot * 2, TB), TB, 0, stream>>>(srcE, dstE, asrc1, adst1, m1, s1, e1, E, Etot);
  edge_agg_kernel<2><<<blk((long long)Etot * 32, TB), TB, 0, stream>>>(srcE, dstE, e1, s1, h1, agg1, E, Etot);
  bias_relu_kernel<<<blk((long long)N * 128, TB), TB, 0, stream>>>(agg1, b1, (long long)N * 128, 127);

  // ---------------- layer 2 (H=1, C=64) ----------------
  gemm_f32_wmma<64><<<dim3(rowGroups), TB, 0, stream>>>(agg1, W2, h2f, N);
  zero_kernel<<<blk((long long)N * 64, TB), TB, 0, stream>>>(agg2, (long long)N * 64);
  zero_kernel<<<blk((long long)N * 2, TB), TB, 0, stream>>>((float*)m2, (long long)N * 2);
  node_alpha_kernel<1><<<blk((long long)N, TB), TB, 0, stream>>>(h2f, as2, ad2, asrc2, adst2, N);
  edge_max_kernel<1><<<blk((long long)Etot, TB), TB, 0, stream>>>(srcE, dstE, asrc2, adst2, m2, E, Etot);
  edge_expsum_kernel<1><<<blk((long long)Etot, TB), TB, 0, stream>>>(srcE, dstE, asrc2, adst2, m2, s2, e2, E, Etot);
  edge_agg_kernel<1><<<blk((long long)Etot * 16, TB), TB, 0, stream>>>(srcE, dstE, e2, s2, h2f, agg2, E, Etot);
  bias_sigmoid_kernel<<<blk((long long)N * 64, TB), TB, 0, stream>>>(agg2, b2, out, (long long)N * 64, 63);
}